// EIRNN_71665824301445
// MI455X (gfx1250) — compile-verified
//
#include <hip/hip_runtime.h>

// EI-RNN on MI455X (gfx1250), fp32 end-to-end via V_WMMA_F32_16X16X4_F32.
//
// Phase 1: xW[t*B+b][h] = x[b][t][:] @ W_xh^T + b_h      (full-chip GEMM -> ws)
// Phase 2: 8 persistent workgroups (16 batch rows each), W_hh resident in LDS
//          (260KB of the 320KB WGP LDS), double-buffered h, 1 barrier/step,
//          xW tile software-pipelined one step ahead + prefetch two ahead.
//          Overwrites xW in place with h_t  (ws stays at T*B*H*4 = 134MB).
// Phase 3: out[b][t][o] = hs[t*B+b][:] @ W_out^T + b_out (full-chip GEMM)

typedef __attribute__((ext_vector_type(2))) float v2f;
typedef __attribute__((ext_vector_type(8))) float v8f;

#define WMMA_F32(a, b, c) \
  __builtin_amdgcn_wmma_f32_16x16x4_f32(false, (a), false, (b), (short)0, (c), false, false)

constexpr int BATCH = 128, TT = 1024, II = 128, HH = 256, OO = 64;
constexpr int RR = BATCH * TT;       // 131072 rows in (t,b) space
constexpr int LDS_STRIDE = 260;      // ≡ 4 (mod 64) -> conflict-free b64 fragment loads

// ---------------- Phase 1: input projection ----------------
// grid (RR/16, 2), block 256 (8 waves). Wave w computes one 16x16 tile.
__global__ void eirnn_xproj(const float* __restrict__ x, const float* __restrict__ Wxh,
                            const float* __restrict__ bh, float* __restrict__ xw) {
  const int lane = threadIdx.x & 31;
  const int wid  = threadIdx.x >> 5;
  const int m    = lane & 15;
  const int hi   = lane >> 4;
  const int r0   = blockIdx.x * 16;
  const int n0   = (blockIdx.y * 8 + wid) * 16;
  const int r    = r0 + m;
  const int t    = r >> 7;       // r / BATCH
  const int b    = r & 127;      // r % BATCH
  const float* arow = x   + (size_t)(b * TT + t) * II + 2 * hi;
  const float* brow = Wxh + (size_t)(n0 + m)     * II + 2 * hi;
  v8f acc = {};
#pragma unroll 8
  for (int k = 0; k < II; k += 4) {
    v2f a  = *(const v2f*)(arow + k);
    v2f bb = *(const v2f*)(brow + k);
    acc = WMMA_F32(a, bb, acc);
  }
  const float bias = bh[n0 + m];
#pragma unroll
  for (int i = 0; i < 8; ++i) {
    const int rout = r0 + i + 8 * hi;               // C layout: M = i + 8*hi, N = m
    xw[(size_t)rout * HH + n0 + m] = acc[i] + bias;
  }
}

// ---------------- Phase 2: recurrence ----------------
// grid 8, block 512 (16 waves). Wave j owns the 16-column H-tile n0 = 16*j.
__global__ __launch_bounds__(512, 1)
void eirnn_recur(const float* __restrict__ Whh, float* __restrict__ xw /* -> hs in place */) {
  extern __shared__ float lds[];
  float* Whh_l = lds;                                   // 256 x 260
  float* h_buf = lds + HH * LDS_STRIDE;                 // 2 x 16 x 260 (double-buffered h)
  const int tid = threadIdx.x;

  // Preload W_hh (row-major, padded stride) with b128 stores.
  for (int idx = tid; idx < (HH * HH) / 4; idx += 512) {
    const int n  = idx >> 6;                            // 64 float4 per row
    const int k4 = idx & 63;
    const float4 v = ((const float4*)Whh)[idx];
    *(float4*)(Whh_l + n * LDS_STRIDE + k4 * 4) = v;    // n*260 ≡ 0 mod 4 -> 16B aligned
  }
  for (int idx = tid; idx < 2 * 16 * LDS_STRIDE; idx += 512) h_buf[idx] = 0.f;  // h0 = 0
  __syncthreads();

  const int lane = tid & 31;
  const int wid  = tid >> 5;        // 0..15 : H output tile
  const int m    = lane & 15;
  const int hi   = lane >> 4;
  const int n0   = wid * 16;
  const int b0   = blockIdx.x * 16;
  const float* b_base = Whh_l + (size_t)(n0 + m) * LDS_STRIDE + 2 * hi;  // W_hh^T frag

  // Software pipeline: xW tile for step t is loaded during step t-1.
  float xwv[8];
#pragma unroll
  for (int i = 0; i < 8; ++i)
    xwv[i] = xw[(size_t)(b0 + i + 8 * hi) * HH + n0 + m];   // t = 0 tile

  int cur = 0;
  for (int t = 0; t < TT; ++t) {
    float xwn[8] = {};
    if (t + 1 < TT) {
#pragma unroll
      for (int i = 0; i < 8; ++i) {
        const int rout = (t + 1) * BATCH + b0 + i + 8 * hi;
        xwn[i] = xw[(size_t)rout * HH + n0 + m];
      }
    }
    if (t + 2 < TT)   // lanes 0..15 cover the 16 rows of the t+2 tile -> global_prefetch_b8
      __builtin_prefetch(xw + (size_t)((t + 2) * BATCH + b0 + m) * HH + n0, 0, 3);

    const float* a_base = h_buf + cur * 16 * LDS_STRIDE + m * LDS_STRIDE + 2 * hi;
    v8f acc = {};
#pragma unroll 8
    for (int k = 0; k < HH; k += 4) {
      v2f a  = *(const v2f*)(a_base + k);
      v2f bb = *(const v2f*)(b_base + k);
      acc = WMMA_F32(a, bb, acc);
    }
    // h_t = relu(xW_t + h_{t-1} @ W_hh^T): write to the other h buffer + global hs.
    float* h_nxt = h_buf + (cur ^ 1) * 16 * LDS_STRIDE;
#pragma unroll
    for (int i = 0; i < 8; ++i) {
      float v = fmaxf(acc[i] + xwv[i], 0.f);
      h_nxt[(i + 8 * hi) * LDS_STRIDE + n0 + m] = v;
      const int rout = t * BATCH + b0 + i + 8 * hi;
      xw[(size_t)rout * HH + n0 + m] = v;               // hs overwrites consumed xW[t]
    }
#pragma unroll
    for (int i = 0; i < 8; ++i) xwv[i] = xwn[i];
    cur ^= 1;
    __syncthreads();   // new h visible; old buffer free for next step's writes
  }
}

// ---------------- Phase 3: output projection ----------------
// grid RR/16, block 128 (4 waves). Wave w computes output tile n0 = 16*w.
__global__ void eirnn_outproj(const float* __restrict__ hs, const float* __restrict__ Wout,
                              const float* __restrict__ bout, float* __restrict__ out) {
  const int lane = threadIdx.x & 31;
  const int wid  = threadIdx.x >> 5;
  const int m    = lane & 15;
  const int hi   = lane >> 4;
  const int r0   = blockIdx.x * 16;
  const int n0   = wid * 16;
  const float* arow = hs   + (size_t)(r0 + m) * HH + 2 * hi;
  const float* brow = Wout + (size_t)(n0 + m) * HH + 2 * hi;
  v8f acc = {};
#pragma unroll 8
  for (int k = 0; k < HH; k += 4) {
    v2f a  = *(const v2f*)(arow + k);
    v2f bb = *(const v2f*)(brow + k);
    acc = WMMA_F32(a, bb, acc);
  }
  const float bias = bout[n0 + m];
#pragma unroll
  for (int i = 0; i < 8; ++i) {
    const int rout = r0 + i + 8 * hi;
    const int t = rout >> 7;
    const int b = rout & 127;
    out[((size_t)b * TT + t) * OO + n0 + m] = acc[i] + bias;
  }
}

extern "C" void kernel_launch(void* const* d_in, const int* in_sizes, int n_in,
                              void* d_out, int out_size, void* d_ws, size_t ws_size,
                              hipStream_t stream) {
  const float* x    = (const float*)d_in[0];  // [B,T,I]
  const float* Wxh  = (const float*)d_in[1];  // [H,I]
  const float* Whh  = (const float*)d_in[2];  // [H,H]
  const float* bh   = (const float*)d_in[3];  // [H]
  const float* Wout = (const float*)d_in[4];  // [O,H]
  const float* bout = (const float*)d_in[5];  // [O]
  float* out  = (float*)d_out;                // [B,T,O]
  float* xwhs = (float*)d_ws;                 // [T*B, H] fp32 = 134.2 MB (xW, then hs in place)

  const size_t ldsBytes = (size_t)(HH * LDS_STRIDE + 2 * 16 * LDS_STRIDE) * sizeof(float); // ~293 KB
  (void)hipFuncSetAttribute((const void*)eirnn_recur,
                            hipFuncAttributeMaxDynamicSharedMemorySize, (int)ldsBytes);

  eirnn_xproj  <<<dim3(RR / 16, 2), 256, 0, stream>>>(x, Wxh, bh, xwhs);
  eirnn_recur  <<<dim3(BATCH / 16), 512, ldsBytes, stream>>>(Whh, xwhs);
  eirnn_outproj<<<dim3(RR / 16), 128, 0, stream>>>(xwhs, Wout, bout, out);
}